// ElectricField_57750130262470
// MI455X (gfx1250) — compile-verified
//
#include <hip/hip_runtime.h>
#include <hip/hip_bf16.h>

// ---------------------------------------------------------------------------
// ElectricField scatter kernel for MI455X (gfx1250).
// HBM-bound: ~154 MB of read-once edge streams -> ~6.6 us floor @ 23.3 TB/s.
// Node arrays (400 KB each) + output (1.2 MB) live in the 192 MB L2, so
// gathers and the 19.2M global_atomic_add_f32 ops are L2-serviced.
// CDNA5-specific paths used: TH=NT nontemporal loads for the edge streams,
// global_prefetch_b8 one stride ahead, no-return f32 global atomics.
// ---------------------------------------------------------------------------

#define BOHR_F 0.52917721067f

__global__ void ef_zero_kernel(float* __restrict__ out, int n) {
    int i = blockIdx.x * blockDim.x + threadIdx.x;
    if (i < n) out[i] = 0.0f;
}

__global__ __launch_bounds__(256) void ef_edge_kernel(
    const int*   __restrict__ edge_src,
    const int*   __restrict__ edge_dst,
    const float* __restrict__ dist,
    const float* __restrict__ vec,
    const float* __restrict__ pol,      // raw polarisability (Angstrom^3)
    const float* __restrict__ charges,
    float*       __restrict__ out,      // [3*N]
    int n_edges)
{
    const int stride = gridDim.x * blockDim.x;
    int e = blockIdx.x * blockDim.x + threadIdx.x;

    // exp(-0.7*t) == exp2(-0.7*log2(e)*t)
    const float kExp = -0.7f * 1.44269504088896340736f;
    const float B2   = BOHR_F * BOHR_F;

    for (; e < n_edges; e += stride) {
        // Prefetch the next tile of the streaming arrays (global_prefetch_b8).
        int ep = e + stride;
        if (ep < n_edges) {
            __builtin_prefetch(dist + ep,         0, 1);
            __builtin_prefetch(vec  + 3 * ep,     0, 1);
            __builtin_prefetch(edge_src + ep,     0, 1);
            __builtin_prefetch(edge_dst + ep,     0, 1);
        }

        // Read-once edge streams: non-temporal so they don't evict the hot
        // node arrays / output from cache.
        int   s  = __builtin_nontemporal_load(edge_src + e);
        int   dn = __builtin_nontemporal_load(edge_dst + e);
        float d  = __builtin_nontemporal_load(dist + e);
        float vx = __builtin_nontemporal_load(vec + 3 * e + 0);
        float vy = __builtin_nontemporal_load(vec + 3 * e + 1);
        float vz = __builtin_nontemporal_load(vec + 3 * e + 2);

        // L2-resident random gathers.
        float ps = pol[s];
        float pd = pol[dn];
        float q  = charges[dn];

        // u = rij / alpha_ij^(1/6); the BOHR factors cancel:
        //   u = dist * (pol_src*pol_dst)^(-1/6)
        float pp = ps * pd;
        float u  = d * __builtin_amdgcn_exp2f(__builtin_amdgcn_logf(pp) * (-1.0f / 6.0f));
        float t  = u * __builtin_amdgcn_sqrtf(u);                 // u^1.5
        float damp = 1.0f - __builtin_amdgcn_exp2f(t * kExp);     // 1 - exp(-0.7*u^1.5)

        // eij = -q_dst * (vec/BOHR) / (dist/BOHR)^3 * damp
        //     = vec * (-q_dst * damp * BOHR^2 / dist^3)
        float sc = -q * damp * B2 * __builtin_amdgcn_rcpf(d * d * d);

        float* o = out + 3 * s;
        atomicAdd(o + 0, sc * vx);   // no-return global_atomic_add_f32
        atomicAdd(o + 1, sc * vy);
        atomicAdd(o + 2, sc * vz);
    }
}

extern "C" void kernel_launch(void* const* d_in, const int* in_sizes, int n_in,
                              void* d_out, int out_size, void* d_ws, size_t ws_size,
                              hipStream_t stream) {
    // setup_inputs() order:
    // 0: species (unused)  1: edge_src  2: edge_dst  3: distances
    // 4: vec [E,3]         5: polarisability          6: charges
    const int*   edge_src = (const int*)  d_in[1];
    const int*   edge_dst = (const int*)  d_in[2];
    const float* dist     = (const float*)d_in[3];
    const float* vec      = (const float*)d_in[4];
    const float* pol      = (const float*)d_in[5];
    const float* charges  = (const float*)d_in[6];
    float*       out      = (float*)d_out;

    const int n_edges = in_sizes[1];

    // 1) zero the accumulator (harness poisons d_out).
    ef_zero_kernel<<<(out_size + 255) / 256, 256, 0, stream>>>(out, out_size);

    // 2) edge scatter. 2048 blocks * 256 threads = 512K lanes,
    //    ~12 grid-stride iterations over 6.4M edges; prefetch covers
    //    the next iteration's cache lines.
    ef_edge_kernel<<<2048, 256, 0, stream>>>(edge_src, edge_dst, dist, vec,
                                             pol, charges, out, n_edges);
}